// FermiNetOneElectronLayer_16767552323746
// MI455X (gfx1250) — compile-verified
//
#include <hip/hip_runtime.h>
#include <hip/hip_bf16.h>
#include <math.h>

// FermiNet 1e layer for MI455X (gfx1250), wave32 + V_WMMA_F32_16X16X4_F32.
// Memory-bound (~805 MB min traffic -> ~35us @ 23.3 TB/s); WMMA keeps the
// 23.6 GFLOP of f32 GEMM work off the VALU so bandwidth stays the limiter.
// Reductions over the 537 MB in_2e stream use b128 loads to minimize VMEM
// instruction issue in the bandwidth-bound phase.

typedef __attribute__((ext_vector_type(2))) float v2f;
typedef __attribute__((ext_vector_type(4))) float v4f;
typedef __attribute__((ext_vector_type(8))) float v8f;

#define STR_A  260   // 16B-aligned rows (260*4 % 16 == 0), bank stride 4
#define STR_G  68
#define STR_C  516

// ---------------------------------------------------------------------------
// Weight repack: W (K x 256) -> WMMA B-fragments.
// pairIdx = kg*512 + ct*32 + lane ; each pair = (W[k][n], W[k+1][n]) with
// n = ct*16 + (lane&15), k = 4*kg + 2*(lane>>4). A wave then loads its
// fragment as one coalesced v2f per lane.
// ---------------------------------------------------------------------------
__device__ inline void pack_w(const float* __restrict__ W, float* __restrict__ out,
                              int pairIdx) {
    int kg   = pairIdx >> 9;
    int rem  = pairIdx & 511;
    int ct   = rem >> 5;
    int lane = rem & 31;
    int n    = ct * 16 + (lane & 15);
    int k    = kg * 4 + ((lane >> 4) << 1);
    v2f v;
    v.x = W[(size_t)k * 256 + n];
    v.y = W[(size_t)(k + 1) * 256 + n];
    ((v2f*)out)[pairIdx] = v;
}

__global__ void fermi_pack(const float* __restrict__ Wu, const float* __restrict__ Wm,
                           const float* __restrict__ W2,
                           float* __restrict__ pu, float* __restrict__ pm,
                           float* __restrict__ p2) {
    int i = blockIdx.x * 256 + threadIdx.x;
    if (i < 32768) {                         // W_unmixed: 64 kg
        pack_w(Wu, pu, i);
    } else if (i < 32768 + 65536) {          // W_mixed: 128 kg
        pack_w(Wm, pm, i - 32768);
    } else if (i < 32768 + 65536 + 8192) {   // W_2e: 16 kg
        pack_w(W2, p2, i - 98304);
    }
}

// ---------------------------------------------------------------------------
// dense_mixed: one block = 8 batches -> 16 rows (8 batches x 2 spins) =
// exactly one WMMA M-tile. K = 512 (cyclic concat of the two spin means).
// ---------------------------------------------------------------------------
__global__ __launch_bounds__(128) void fermi_mixed(const float* __restrict__ in_1e,
                                                   const float* __restrict__ pWm,
                                                   float* __restrict__ dm) {
    __shared__ float cm[16 * STR_C];
    const int tid  = threadIdx.x;
    const int lane = tid & 31;
    const int wave = tid >> 5;
    const int b0   = blockIdx.x * 8;

    // Spin means of in_1e, vectorized: one quad of 4 dims per (batch, spin)
    // slot. 1024 quads total, 8 passes of 128 threads, b128 loads.
    for (int idx = tid; idx < 8 * 2 * 64; idx += 128) {
        int g = idx >> 7, s = (idx >> 6) & 1, q = idx & 63;
        const v4f* p = (const v4f*)(in_1e + ((size_t)(b0 + g) * 32 + s * 16) * 256) + q;
        v4f sum = {};
        #pragma unroll
        for (int i = 0; i < 16; ++i) sum += p[i * 64];   // row stride 256 f = 64 quads
        sum *= 0.0625f;
        *(v4f*)&cm[(2 * g + s) * STR_C + q * 4]           = sum;  // own-spin 1st half
        *(v4f*)&cm[(2 * g + 1 - s) * STR_C + 256 + q * 4] = sum;  // other row 2nd half
    }
    __syncthreads();

    v8f acc[4] = {};
    const int row = lane & 15;
    const int kb  = (lane >> 4) << 1;
    const v2f* pb = (const v2f*)pWm;

    for (int kg = 0; kg < 128; ++kg) {
        v2f a = *(const v2f*)&cm[row * STR_C + kg * 4 + kb];
        int base = kg * 512 + wave * 128 + lane;
        #pragma unroll
        for (int t = 0; t < 4; ++t) {
            v2f b = pb[base + t * 32];
            acc[t] = __builtin_amdgcn_wmma_f32_16x16x4_f32(
                false, a, false, b, (short)0, acc[t], false, false);
        }
    }

    const int mh = (lane >> 4) << 3;
    #pragma unroll
    for (int t = 0; t < 4; ++t) {
        int n = wave * 64 + t * 16 + (lane & 15);
        #pragma unroll
        for (int r = 0; r < 8; ++r) {
            int m = r + mh;          // row = 2*g + s
            int g = m >> 1, s = m & 1;
            dm[((size_t)(b0 + g) * 2 + s) * 256 + n] = acc[t][r];
        }
    }
}

// ---------------------------------------------------------------------------
// Main fused kernel: one block = one (batch, spin) group of 16 particles.
//   - stage in_1e 16x256 tile in LDS (A-fragments + residual)
//   - reduce in_2e slice -> concat'd g2 16x64 tile in LDS (single HBM pass)
//   - WMMA: K=256 over W_unmixed, K=64 over W_2e, 16x64 per wave
//   - epilogue: + bias + dense_mixed, tanh, + residual
// ---------------------------------------------------------------------------
__global__ __launch_bounds__(128) void fermi_main(const float* __restrict__ in_1e,
                                                  const float* __restrict__ in_2e,
                                                  const float* __restrict__ bias,
                                                  const float* __restrict__ dm,
                                                  const float* __restrict__ pWu,
                                                  const float* __restrict__ pW2,
                                                  float* __restrict__ out) {
    __shared__ float a_tile[16 * STR_A];
    __shared__ float g2s[16 * STR_G];
    const int tid  = threadIdx.x;
    const int lane = tid & 31;
    const int wave = tid >> 5;
    const int r0   = blockIdx.x * 16;     // first global row (b*32 + i)
    const int b    = r0 >> 5;
    const int spin = (r0 >> 4) & 1;

    // Stage in_1e tile (b128, coalesced).
    for (int idx = tid; idx < 16 * 64; idx += 128) {
        int row = idx >> 6, c4 = idx & 63;
        v4f v = ((const v4f*)in_1e)[(size_t)(r0 + row) * 64 + c4];
        *(v4f*)&a_tile[row * STR_A + c4 * 4] = v;
    }

    // in_2e block means -> cyclic-concat'd g2 (16 rows x 64). Each thread
    // owns one quad of 4 d-values for one row i: exactly 128 slots, b128
    // loads, 32 loads/thread over the 64 KB slice.
    {
        const int i  = tid >> 3;          // 0..15
        const int dq = tid & 7;           // d-quad 0..7 (32 dims = 8 quads)
        const v4f* p = (const v4f*)(in_2e + ((size_t)(r0 + i) * 32) * 32) + dq;
        v4f s0 = {}, s1 = {};
        #pragma unroll
        for (int j = 0; j < 16; ++j) s0 += p[j * 8];    // j row = 32 f = 8 quads
        #pragma unroll
        for (int j = 16; j < 32; ++j) s1 += p[j * 8];
        s0 *= 0.0625f; s1 *= 0.0625f;
        if (spin == 0) {
            *(v4f*)&g2s[i * STR_G + dq * 4]      = s0;
            *(v4f*)&g2s[i * STR_G + 32 + dq * 4] = s1;
        } else {
            *(v4f*)&g2s[i * STR_G + dq * 4]      = s1;
            *(v4f*)&g2s[i * STR_G + 32 + dq * 4] = s0;
        }
    }
    __syncthreads();

    v8f acc[4] = {};
    const int row = lane & 15;
    const int kb  = (lane >> 4) << 1;
    const v2f* pu = (const v2f*)pWu;
    const v2f* p2 = (const v2f*)pW2;

    // K = 256 over W_unmixed.
    for (int kg = 0; kg < 64; ++kg) {
        v2f a = *(const v2f*)&a_tile[row * STR_A + kg * 4 + kb];
        int base = kg * 512 + wave * 128 + lane;
        #pragma unroll
        for (int t = 0; t < 4; ++t) {
            v2f bf = pu[base + t * 32];
            acc[t] = __builtin_amdgcn_wmma_f32_16x16x4_f32(
                false, a, false, bf, (short)0, acc[t], false, false);
        }
    }
    // K = 64 over W_2e (fused dense_2e).
    for (int kg = 0; kg < 16; ++kg) {
        v2f a = *(const v2f*)&g2s[row * STR_G + kg * 4 + kb];
        int base = kg * 512 + wave * 128 + lane;
        #pragma unroll
        for (int t = 0; t < 4; ++t) {
            v2f bf = p2[base + t * 32];
            acc[t] = __builtin_amdgcn_wmma_f32_16x16x4_f32(
                false, a, false, bf, (short)0, acc[t], false, false);
        }
    }

    // Epilogue: bias + dense_mixed (constant over the tile's rows), tanh,
    // residual from staged in_1e.
    const float* dmv = dm + ((size_t)b * 2 + spin) * 256;
    const int mh = (lane >> 4) << 3;
    #pragma unroll
    for (int t = 0; t < 4; ++t) {
        int n = wave * 64 + t * 16 + (lane & 15);
        float add = bias[n] + dmv[n];
        #pragma unroll
        for (int r = 0; r < 8; ++r) {
            int m = r + mh;
            float val = tanhf(acc[t][r] + add) + a_tile[m * STR_A + n];
            out[(size_t)(r0 + m) * 256 + n] = val;
        }
    }
}

// ---------------------------------------------------------------------------
extern "C" void kernel_launch(void* const* d_in, const int* in_sizes, int n_in,
                              void* d_out, int out_size, void* d_ws, size_t ws_size,
                              hipStream_t stream) {
    const float* in_1e = (const float*)d_in[0];   // (4096, 32, 256)
    const float* in_2e = (const float*)d_in[1];   // (4096, 32, 32, 32)
    const float* Wu    = (const float*)d_in[2];   // (256, 256)
    const float* bias  = (const float*)d_in[3];   // (256,)
    const float* Wm    = (const float*)d_in[4];   // (512, 256)
    const float* W2    = (const float*)d_in[5];   // (64, 256)
    float* out = (float*)d_out;

    float* ws = (float*)d_ws;
    float* dm = ws;                    // 4096*2*256      = 2,097,152 floats
    float* pu = dm + 2097152;          // packed W_unmixed:     65,536
    float* pm = pu + 65536;            // packed W_mixed:      131,072
    float* p2 = pm + 131072;           // packed W_2e:          16,384
    (void)in_sizes; (void)n_in; (void)out_size; (void)ws_size;

    fermi_pack <<<416, 256, 0, stream>>>(Wu, Wm, W2, pu, pm, p2);
    fermi_mixed<<<512, 128, 0, stream>>>(in_1e, pm, dm);
    fermi_main <<<8192, 128, 0, stream>>>(in_1e, in_2e, bias, dm, pu, p2, out);
}